// RagContrastiveLoss_2886218023668
// MI455X (gfx1250) — compile-verified
//
#include <hip/hip_runtime.h>
#include <hip/hip_bf16.h>

// RAG contrastive loss for MI455X (gfx1250, wave32).
//
// Pipeline (all on `stream`, graph-capture safe):
//   1) zero_ws            : zero counts/sums/accumulators
//   2) seg_accum          : per-pixel scatter into per-segment count+sum (native f32 atomics)
//   3) finalize_means     : sums /= count (in place), inv_n = 1/count
//   4) intra_loss         : stream 128MB embeddings (L2-resident after pass 2),
//                           hinge(||mean[seg]-emb||-0.5)*inv_n, hierarchical reduce
//   5) inter_loss_wmma    : 16 edges per wave; diff matrix D (16x32 f32) fed to
//                           v_wmma_f32_16x16x4_f32 as both A and B -> Gram D*D^T,
//                           diagonal = squared distances; hinge(1.5 - d)
//   6) finalize_out       : out = inter/n_edges + intra/C

#define CSEG 4096
#define EDIM 32

// workspace layout in floats
#define WS_CNT   0
#define WS_SUMS  (CSEG)                    // reused in-place as means
#define WS_MEANS (WS_SUMS)
#define WS_INVN  (CSEG + CSEG*EDIM)
#define WS_ACC   (2*CSEG + CSEG*EDIM)      // [0]=intra sum, [1]=inter sum
#define WS_TOTAL (WS_ACC + 2)

typedef __attribute__((ext_vector_type(2))) float v2f;
typedef __attribute__((ext_vector_type(8))) float v8f;

__device__ __forceinline__ void atomicAddF(float* p, float v) {
    // lowers to global_atomic_add_f32 (no-return) with relaxed/agent semantics
    __hip_atomic_fetch_add(p, v, __ATOMIC_RELAXED, __HIP_MEMORY_SCOPE_AGENT);
}

__device__ __forceinline__ float block_reduce_sum(float v) {
    #pragma unroll
    for (int o = 16; o > 0; o >>= 1) v += __shfl_down(v, o, 32);
    __shared__ float smem[8];
    const int lane = threadIdx.x & 31;
    const int wid  = threadIdx.x >> 5;
    if (lane == 0) smem[wid] = v;
    __syncthreads();
    if (wid == 0) {
        const int nw = blockDim.x >> 5;
        v = (lane < nw) ? smem[lane] : 0.0f;
        #pragma unroll
        for (int o = 4; o > 0; o >>= 1) v += __shfl_down(v, o, 32);
    }
    return v;  // valid on thread 0
}

__global__ void zero_ws(float* __restrict__ ws, int n) {
    int i = blockIdx.x * blockDim.x + threadIdx.x;
    if (i < n) ws[i] = 0.0f;
}

// Pass 1: per-segment pixel count + embedding sum. emb is channel-major (E,N):
// for fixed e, consecutive lanes hit consecutive addresses (coalesced stream).
__global__ void seg_accum(const float* __restrict__ emb, const int* __restrict__ seg,
                          float* __restrict__ ws, int N) {
    int p = blockIdx.x * blockDim.x + threadIdx.x;
    if (p >= N) return;
    const int s = seg[p];
    atomicAddF(ws + WS_CNT + s, 1.0f);
    float* row = ws + WS_SUMS + (size_t)s * EDIM;
    #pragma unroll
    for (int e = 0; e < EDIM; ++e) {
        atomicAddF(row + e, emb[(size_t)e * N + p]);
    }
}

__global__ void finalize_means(float* __restrict__ ws) {
    int i = blockIdx.x * blockDim.x + threadIdx.x;
    if (i >= CSEG * EDIM) return;
    const int c = i >> 5;
    const float n = ws[WS_CNT + c];
    ws[WS_SUMS + i] = ws[WS_SUMS + i] / n;     // in-place: sums -> means
    if ((i & (EDIM - 1)) == 0) ws[WS_INVN + c] = 1.0f / n;
}

// Pass 2: intra term. Embeddings streamed (should be L2-hot after pass 1:
// 128MB data vs 192MB L2); means row gathered as 8x float4 from 512KB table.
__global__ void intra_loss(const float* __restrict__ emb, const int* __restrict__ seg,
                           const float* __restrict__ ws, float* __restrict__ acc, int N) {
    const float* means = ws + WS_MEANS;
    const float* invn  = ws + WS_INVN;
    float local = 0.0f;
    for (int p = blockIdx.x * blockDim.x + threadIdx.x; p < N;
         p += gridDim.x * blockDim.x) {
        const int s = seg[p];
        const float4* mrow = reinterpret_cast<const float4*>(means + (size_t)s * EDIM);
        float d2 = 0.0f;
        #pragma unroll
        for (int q = 0; q < 8; ++q) {
            const float4 m = mrow[q];
            const float x0 = emb[(size_t)(4 * q + 0) * N + p];
            const float x1 = emb[(size_t)(4 * q + 1) * N + p];
            const float x2 = emb[(size_t)(4 * q + 2) * N + p];
            const float x3 = emb[(size_t)(4 * q + 3) * N + p];
            const float t0 = m.x - x0, t1 = m.y - x1, t2 = m.z - x2, t3 = m.w - x3;
            d2 = fmaf(t0, t0, d2); d2 = fmaf(t1, t1, d2);
            d2 = fmaf(t2, t2, d2); d2 = fmaf(t3, t3, d2);
        }
        float h = sqrtf(d2) - 0.5f;
        h = (h > 0.0f) ? h : 0.0f;
        local += h * invn[s];
    }
    const float bs = block_reduce_sum(local);
    if (threadIdx.x == 0) atomicAddF(acc + 0, bs);
}

// Pass 3: inter term via WMMA. Each wave owns 16 edges; per K-chunk of 4,
// lane l (l<16) holds D[l][k0+0..1], lane l+16 holds D[l-16][k0+2..3] — the
// documented 16x4 f32 A layout. The B layout is its N<->M mirror, so passing
// the same registers as A and B yields Gram = D * D^T; diag = squared dists.
__global__ void inter_loss_wmma(const int* __restrict__ edges,
                                const float* __restrict__ ws,
                                float* __restrict__ acc, int n_edges) {
    const float* means = ws + WS_MEANS;
    const int lane = threadIdx.x & 31;
    const int wave = (blockIdx.x * blockDim.x + threadIdx.x) >> 5;
    const int base = wave * 16;                 // wave-uniform
    float h = 0.0f;
    if (base < n_edges) {                       // uniform branch: EXEC stays full
        const int r     = lane & 15;            // edge row 0..15 within wave
        const int khalf = (lane >> 4) << 1;     // 0 (lanes 0-15) or 2 (lanes 16-31)
        const int ei    = base + r;
        const bool valid = (ei < n_edges);
        const int ec    = valid ? ei : (n_edges - 1);   // clamp to keep EXEC full
        const int u = edges[ec];
        const int v = edges[n_edges + ec];
        const float2* mu2 = reinterpret_cast<const float2*>(means + (size_t)u * EDIM);
        const float2* mv2 = reinterpret_cast<const float2*>(means + (size_t)v * EDIM);
        v8f c = {};
        #pragma unroll
        for (int k0 = 0; k0 < EDIM; k0 += 4) {
            const float2 pu = mu2[(k0 + khalf) >> 1];
            const float2 pv = mv2[(k0 + khalf) >> 1];
            v2f a;
            a.x = pu.x - pv.x;
            a.y = pu.y - pv.y;
            // D = A * B + C : A=B=diff chunk -> accumulate Gram matrix
            c = __builtin_amdgcn_wmma_f32_16x16x4_f32(
                    false, a, false, a, (short)0, c, false, false);
        }
        // Extract diagonal of the 16x16 Gram (C layout: VGPR v holds rows M=v
        // on lanes 0-15 and M=v+8 on lanes 16-31, N=lane&15).
        float d2 = 0.0f;
        #pragma unroll
        for (int vv = 0; vv < 8; ++vv) {
            const float x = __shfl(c[vv], vv, 32);        // diag m = vv
            const float y = __shfl(c[vv], vv + 24, 32);   // diag m = vv + 8
            if (lane == vv)     d2 = x;
            if (lane == vv + 8) d2 = y;
        }
        if (lane < 16 && valid) {
            const float d = sqrtf(d2 > 0.0f ? d2 : 0.0f);
            const float t = 1.5f - d;
            h = (t > 0.0f) ? t : 0.0f;
        }
    }
    const float bs = block_reduce_sum(h);
    if (threadIdx.x == 0) atomicAddF(acc + 1, bs);
}

__global__ void finalize_out(const float* __restrict__ acc, float* __restrict__ out,
                             float inv_edges) {
    if (blockIdx.x == 0 && threadIdx.x == 0) {
        out[0] = acc[1] * inv_edges + acc[0] * (1.0f / (float)CSEG);
    }
}

extern "C" void kernel_launch(void* const* d_in, const int* in_sizes, int n_in,
                              void* d_out, int out_size, void* d_ws, size_t ws_size,
                              hipStream_t stream) {
    const float* emb   = (const float*)d_in[0];   // (1,32,1,1024,1024) f32, channel-major
    const int*   seg   = (const int*)d_in[1];     // (1,1,1,1024,1024) i32
    const int*   edges = (const int*)d_in[2];     // (2, n_edges) i32
    float* out = (float*)d_out;
    float* ws  = (float*)d_ws;

    const int N       = in_sizes[1];       // 1024*1024 pixels
    const int n_edges = in_sizes[2] / 2;   // 16384

    zero_ws<<<(WS_TOTAL + 255) / 256, 256, 0, stream>>>(ws, WS_TOTAL);

    seg_accum<<<(N + 255) / 256, 256, 0, stream>>>(emb, seg, ws, N);

    finalize_means<<<(CSEG * EDIM + 255) / 256, 256, 0, stream>>>(ws);

    intra_loss<<<1024, 256, 0, stream>>>(emb, seg, ws, ws + WS_ACC, N);

    const int waves  = (n_edges + 15) / 16;
    const int blocks = (waves * 32 + 255) / 256;
    inter_loss_wmma<<<blocks, 256, 0, stream>>>(edges, ws, ws + WS_ACC, n_edges);

    finalize_out<<<1, 64, 0, stream>>>(ws + WS_ACC, out, 1.0f / (float)n_edges);
}